// Convention_33182917329119
// MI455X (gfx1250) — compile-verified
//
#include <hip/hip_runtime.h>
#include <math.h>

// Reference constants
#define DT_C     0.05f
#define N_STEPS  41

typedef __attribute__((ext_vector_type(2))) float v2f;
typedef __attribute__((ext_vector_type(8))) float v8f;

// ---------------------------------------------------------------------------
// tanh: gfx1250 has V_TANH_F32 (TRANS32, co-executes with VALU). Prefer the
// hardware unit; fall back to libm tanhf if the builtin name differs.
// ---------------------------------------------------------------------------
__device__ __forceinline__ float fast_tanhf(float x) {
#if __has_builtin(__builtin_amdgcn_tanhf)
  return __builtin_amdgcn_tanhf(x);
#elif __has_builtin(__builtin_amdgcn_tanh_f32)
  return __builtin_amdgcn_tanh_f32(x);
#else
  return tanhf(x);
#endif
}

// ---------------------------------------------------------------------------
// Exact 32-lane wave sum via V_WMMA_F32_16X16X4_F32.
//   A: V0 = V1 = x  =>  A[m][0..1] = x(lane m), A[m][2..3] = x(lane m+16)
//      (documented 32-bit 16x4 A layout)
//   B: all elements 0.5 (layout-independent since constant)
//   => D[m][n] = 0.5*(2*x(m) + 2*x(m+16)) = x(m) + x(m+16)
//   C/D layout: VGPR r holds row r (lanes 0-15) and row r+8 (lanes 16-31),
//   so sum(c[0..7]) gives sum_{m=0..7} pairs in lanes 0-15 and
//   sum_{m=8..15} pairs in lanes 16-31; one xor-16 shuffle finishes it.
// Must be called with EXEC all ones (all callers are convergent full waves).
// ---------------------------------------------------------------------------
__device__ __forceinline__ float wave_sum32(float x) {
#if __has_builtin(__builtin_amdgcn_wmma_f32_16x16x4_f32)
  v2f a; a.x = x;    a.y = x;
  v2f b; b.x = 0.5f; b.y = 0.5f;
  v8f c = {};
  c = __builtin_amdgcn_wmma_f32_16x16x4_f32(
      /*neg_a=*/false, a, /*neg_b=*/false, b,
      /*c_mod=*/(short)0, c, /*reuse_a=*/false, /*reuse_b=*/false);
  float p = ((c[0] + c[1]) + (c[2] + c[3])) + ((c[4] + c[5]) + (c[6] + c[7]));
  p += __shfl_xor(p, 16, 32);
  return p;
#else
  for (int off = 16; off > 0; off >>= 1) x += __shfl_xor(x, off, 32);
  return x;
#endif
}

// ---------------------------------------------------------------------------
// One task per lane: 40-step rollout fully in registers. Weights are uniform
// -> scalar loads -> SGPRs. Deterministic per-block partial sums (no atomics).
// ---------------------------------------------------------------------------
__global__ __launch_bounds__(256) void rollout_kernel(
    const float* __restrict__ omega,
    const float* __restrict__ Wh1, const float* __restrict__ bh1,
    const float* __restrict__ Wh2, const float* __restrict__ bh2,
    const float* __restrict__ Wr1, const float* __restrict__ br1,
    const float* __restrict__ Wr2, const float* __restrict__ br2,
    int n, float* __restrict__ blockU, float* __restrict__ blockV)
{
  const int tid   = blockIdx.x * blockDim.x + threadIdx.x;
  const int t     = (tid < n) ? tid : (n - 1);
  const float val = (tid < n) ? 1.0f : 0.0f;

  // Uniform weights -> scalar registers after unrolled constant indexing.
  float wh1[16], vbh1[4], wh2[4], wr1[9], vbr1[3], wr2[3];
#pragma unroll
  for (int i = 0; i < 16; ++i) wh1[i] = Wh1[i];
#pragma unroll
  for (int i = 0; i < 4; ++i)  vbh1[i] = bh1[i];
#pragma unroll
  for (int i = 0; i < 4; ++i)  wh2[i] = Wh2[i];
#pragma unroll
  for (int i = 0; i < 9; ++i)  wr1[i] = Wr1[i];
#pragma unroll
  for (int i = 0; i < 3; ++i)  vbr1[i] = br1[i];
#pragma unroll
  for (int i = 0; i < 3; ++i)  wr2[i] = Wr2[i];
  const float bh2s = bh2[0];
  const float br2s = br2[0];

  const float ss0 = omega[t];       // omega[0, t]
  const float ss1 = omega[n + t];   // omega[1, t]

  float s0 = 0.0f, s1 = 0.0f;
  float err = 0.0f, eff = 0.0f, nor = 0.0f;

  for (int k = 0; k < N_STEPS - 1; ++k) {
    const float e0 = ss0 - s0;
    const float e1 = ss1 - s1;
    err += 10.0f * e0 * e0 + e1 * e1;
    const float zt = e0 + 0.1f * e1;

    // human MLP: xh = [ss0, ss1, s0, s1]; tanh on both layers
    const float h0 = fast_tanhf(wh1[0]*ss0  + wh1[1]*ss1  + wh1[2]*s0  + wh1[3]*s1  + vbh1[0]);
    const float h1 = fast_tanhf(wh1[4]*ss0  + wh1[5]*ss1  + wh1[6]*s0  + wh1[7]*s1  + vbh1[1]);
    const float h2 = fast_tanhf(wh1[8]*ss0  + wh1[9]*ss1  + wh1[10]*s0 + wh1[11]*s1 + vbh1[2]);
    const float h3 = fast_tanhf(wh1[12]*ss0 + wh1[13]*ss1 + wh1[14]*s0 + wh1[15]*s1 + vbh1[3]);
    const float z  = fast_tanhf(wh2[0]*h0 + wh2[1]*h1 + wh2[2]*h2 + wh2[3]*h3 + bh2s);

    // robot MLP: xr = [s0, s1, z]; tanh hidden, linear out
    const float r0 = fast_tanhf(wr1[0]*s0 + wr1[1]*s1 + wr1[2]*z + vbr1[0]);
    const float r1 = fast_tanhf(wr1[3]*s0 + wr1[4]*s1 + wr1[5]*z + vbr1[1]);
    const float r2 = fast_tanhf(wr1[6]*s0 + wr1[7]*s1 + wr1[8]*z + vbr1[2]);
    const float aa = wr2[0]*r0 + wr2[1]*r1 + wr2[2]*r2 + br2s;

    // s_next = A @ s + B * a (uses old s1 for s0 update)
    const float ns0 = fmaf(DT_C, s1, s0);
    const float ns1 = fmaf(DT_C, aa, s1);
    s0 = ns0; s1 = ns1;

    eff += (fabsf(z) > 0.01f) ? 1.0f : 0.0f;
    const float d = zt - z;
    nor += d * d;
  }
  // final error term
  {
    const float e0 = ss0 - s0;
    const float e1 = ss1 - s1;
    err += 10.0f * e0 * e0 + e1 * e1;
  }

  // Wave-exact sums via WMMA (full EXEC: every thread reaches here).
  const float wu = wave_sum32((err + eff) * val);
  const float wv = wave_sum32(nor * val);

  __shared__ float su[8], sv[8];
  const int wave = threadIdx.x >> 5;
  const int lane = threadIdx.x & 31;
  if (lane == 0) { su[wave] = wu; sv[wave] = wv; }
  __syncthreads();
  if (threadIdx.x == 0) {
    float bu = 0.0f, bv = 0.0f;
#pragma unroll
    for (int i = 0; i < 8; ++i) { bu += su[i]; bv += sv[i]; }
    blockU[blockIdx.x] = bu;
    blockV[blockIdx.x] = bv;
  }
}

// ---------------------------------------------------------------------------
// Deterministic single-block finalize: fixed strided order + fixed LDS tree.
// ---------------------------------------------------------------------------
__global__ __launch_bounds__(256) void finalize_kernel(
    const float* __restrict__ blockU, const float* __restrict__ blockV,
    int nblocks, int n, const float* __restrict__ alpha,
    float* __restrict__ out)
{
  __shared__ float su[256], sv[256];
  float u = 0.0f, v = 0.0f;
  for (int i = threadIdx.x; i < nblocks; i += 256) { u += blockU[i]; v += blockV[i]; }
  su[threadIdx.x] = u; sv[threadIdx.x] = v;
  __syncthreads();
  for (int s = 128; s > 0; s >>= 1) {
    if (threadIdx.x < s) {
      su[threadIdx.x] += su[threadIdx.x + s];
      sv[threadIdx.x] += sv[threadIdx.x + s];
    }
    __syncthreads();
  }
  if (threadIdx.x == 0) {
    const float inv = 1.0f / (float)n;
    out[0] = su[0] * inv + alpha[0] * (sv[0] * inv);
  }
}

extern "C" void kernel_launch(void* const* d_in, const int* in_sizes, int n_in,
                              void* d_out, int out_size, void* d_ws, size_t ws_size,
                              hipStream_t stream) {
  (void)n_in; (void)out_size; (void)ws_size;
  const float* omega = (const float*)d_in[0];
  const float* Wh1   = (const float*)d_in[1];
  const float* bh1   = (const float*)d_in[2];
  const float* Wh2   = (const float*)d_in[3];
  const float* bh2   = (const float*)d_in[4];
  const float* Wr1   = (const float*)d_in[5];
  const float* br1   = (const float*)d_in[6];
  const float* Wr2   = (const float*)d_in[7];
  const float* br2   = (const float*)d_in[8];
  const float* alpha = (const float*)d_in[9];

  const int n       = in_sizes[0] / 2;          // omega is [2, N]
  const int nblocks = (n + 255) / 256;

  float* blockU = (float*)d_ws;                 // nblocks floats
  float* blockV = blockU + nblocks;             // nblocks floats

  rollout_kernel<<<nblocks, 256, 0, stream>>>(omega, Wh1, bh1, Wh2, bh2,
                                              Wr1, br1, Wr2, br2,
                                              n, blockU, blockV);
  finalize_kernel<<<1, 256, 0, stream>>>(blockU, blockV, nblocks, n, alpha,
                                         (float*)d_out);
}